// GATLayer_71554155152283
// MI455X (gfx1250) — compile-verified
//
#include <hip/hip_runtime.h>
#include <hip/hip_bf16.h>

#define CIN 128
#define HEADS 8
#define CH 8            // Cout per head
#define HC (HEADS * CH) // 64
#define NEG_SLOPE 0.2f
#define SM_EPS 1e-16f

typedef __attribute__((ext_vector_type(2))) float v2f;
typedef __attribute__((ext_vector_type(8))) float v8f;

// ---------------------------------------------------------------------------
// K1: h[N,64] = x[N,128] @ W[128,64]  using V_WMMA_F32_16X16X4_F32 (fp32 WMMA)
// One wave per 16x16 output tile; 4 waves/block cover the 64 output columns.
// A 16x4 layout: lane m=L%16; VGPR0/1 hold K={0,1} (lanes 0-15) / K={2,3}
// (lanes 16-31). B 4x16 / C 16x16 are row-striped across lanes per ISA 7.12.2.
// ---------------------------------------------------------------------------
__global__ __launch_bounds__(128) void k_gemm(const float* __restrict__ x,
                                              const float* __restrict__ w,
                                              float* __restrict__ h, int N) {
  const int lane = threadIdx.x & 31;
  const int wv   = threadIdx.x >> 5;        // 0..3 -> column tile
  const int row0 = blockIdx.x * 16;
  const int col0 = wv * 16;
  const int l16  = lane & 15;
  const int kh   = (lane >> 4) << 1;        // 0 or 2 (upper half-wave = K+2)

  int r = row0 + l16;
  if (r >= N) r = N - 1;                    // clamp loads: EXEC must stay full for WMMA
  const float* xr = x + (size_t)r * CIN;
  const float* wc = w + col0 + l16;

  v8f acc = {};
  #pragma unroll 4
  for (int k = 0; k < CIN; k += 4) {
    v2f a, b;
    a.x = xr[k + kh + 0];
    a.y = xr[k + kh + 1];
    b.x = wc[(size_t)(k + kh + 0) * HC];
    b.y = wc[(size_t)(k + kh + 1) * HC];
    acc = __builtin_amdgcn_wmma_f32_16x16x4_f32(false, a, false, b,
                                                (short)0, acc, false, false);
  }

  // C/D layout: VGPR v -> M = v (lanes 0-15) or v+8 (lanes 16-31), N = lane%16
  const int mb = row0 + ((lane >> 4) << 3);
  float* o = h + (size_t)mb * HC + col0 + l16;
  if (row0 + 16 <= N) {
    // full tile (always the case when N % 16 == 0): unguarded clause of stores
    #pragma unroll
    for (int v = 0; v < 8; ++v) o[(size_t)v * HC] = acc[v];
  } else {
    #pragma unroll
    for (int v = 0; v < 8; ++v)
      if (mb + v < N) o[(size_t)v * HC] = acc[v];
  }
}

// ---------------------------------------------------------------------------
// K2: per (node, head): s = h . att_src, t = h . att_tgt; init amax/asum;
//     zero the output accumulator (d_out is poisoned by the harness).
// ---------------------------------------------------------------------------
__global__ void k_node_prep(const float* __restrict__ h, const float* __restrict__ att,
                            float* __restrict__ s, float* __restrict__ t,
                            float* __restrict__ amax, float* __restrict__ asum,
                            float* __restrict__ out, int NH) {
  int idx = blockIdx.x * blockDim.x + threadIdx.x;   // idx = n*HEADS + hh
  if (idx >= NH) return;
  int hh = idx & (HEADS - 1);
  const float* hr = h + (size_t)idx * CH;            // h[n][hh][0..7]
  const float* as = att + hh * (2 * CH);
  float* o = out + (size_t)idx * CH;
  float sv = 0.f, tv = 0.f;
  #pragma unroll
  for (int c = 0; c < CH; ++c) {
    float hv = hr[c];
    sv += hv * as[c];
    tv += hv * as[CH + c];
    o[c] = 0.f;
  }
  s[idx] = sv;
  t[idx] = tv;
  amax[idx] = -__builtin_inff();
  asum[idx] = 0.f;
}

// float atomic max via signed-max / unsigned-min split (monotone in float
// order, valid with -inf init and mixed sign interleaving).
__device__ inline void atomicMaxF(float* addr, float val) {
  unsigned u = __float_as_uint(val);
  if (u >> 31) atomicMin((unsigned int*)addr, u);
  else         atomicMax((int*)addr, (int)u);
}

// ---------------------------------------------------------------------------
// K3: per (edge, head): leaky-relu logit, store alpha, segment-max via atomics
// ---------------------------------------------------------------------------
__global__ void k_edge_alpha(const long long* __restrict__ ei,
                             const float* __restrict__ s, const float* __restrict__ t,
                             float* __restrict__ alpha, float* __restrict__ amax,
                             int E) {
  int idx = blockIdx.x * blockDim.x + threadIdx.x;
  if (idx >= E * HEADS) return;
  int e  = idx >> 3;
  int hh = idx & 7;
  int src = (int)ei[e];
  int tgt = (int)ei[(size_t)E + e];
  float a = s[src * HEADS + hh] + t[tgt * HEADS + hh];
  a = (a >= 0.f) ? a : NEG_SLOPE * a;
  alpha[idx] = a;
  atomicMaxF(&amax[tgt * HEADS + hh], a);
}

// ---------------------------------------------------------------------------
// K4: per (edge, head): wgt = exp(alpha - amax[tgt]); accumulate denominator
//     and the UNnormalized message sum (normalize per-node in K5 -> saves a
//     whole edge pass). 8 lanes share one edge -> coalesced 32B h[src] reads.
// ---------------------------------------------------------------------------
__global__ void k_edge_agg(const long long* __restrict__ ei,
                           const float* __restrict__ alpha,
                           const float* __restrict__ amax,
                           const float* __restrict__ h,
                           float* __restrict__ asum, float* __restrict__ out,
                           int E) {
  int idx = blockIdx.x * blockDim.x + threadIdx.x;
  if (idx >= E * HEADS) return;
  int e  = idx >> 3;
  int hh = idx & 7;
  int src = (int)ei[e];
  int tgt = (int)ei[(size_t)E + e];
  int th  = tgt * HEADS + hh;
  float wgt = __expf(alpha[idx] - amax[th]);
  atomicAdd(&asum[th], wgt);
  const float* hs = h + ((size_t)src * HEADS + hh) * CH;
  float* o = out + ((size_t)tgt * HEADS + hh) * CH;
  #pragma unroll
  for (int c = 0; c < CH; ++c) atomicAdd(o + c, wgt * hs[c]);
}

// ---------------------------------------------------------------------------
// K5: out = out / max(asum, eps) + bias
// ---------------------------------------------------------------------------
__global__ void k_finalize(float* __restrict__ out, const float* __restrict__ asum,
                           const float* __restrict__ bias, int total) {
  int idx = blockIdx.x * blockDim.x + threadIdx.x;   // n*64 + hh*8 + c
  if (idx >= total) return;
  float d = asum[idx >> 3];                          // (n*HEADS + hh)
  d = (d > SM_EPS) ? d : SM_EPS;
  out[idx] = out[idx] / d + bias[idx & (HC - 1)];
}

extern "C" void kernel_launch(void* const* d_in, const int* in_sizes, int n_in,
                              void* d_out, int out_size, void* d_ws, size_t ws_size,
                              hipStream_t stream) {
  const float*     x    = (const float*)d_in[0];
  const long long* ei   = (const long long*)d_in[1];
  const float*     w    = (const float*)d_in[2];
  const float*     att  = (const float*)d_in[3];
  const float*     bias = (const float*)d_in[4];
  float*           out  = (float*)d_out;

  const int N = in_sizes[0] / CIN;   // 100000
  const int E = in_sizes[1] / 2;     // 1600000

  // workspace layout (floats): h[N*64] | s[N*8] | t[N*8] | amax[N*8] | asum[N*8] | alpha[E*8]
  float* hbuf  = (float*)d_ws;
  float* sbuf  = hbuf + (size_t)N * HC;
  float* tbuf  = sbuf + (size_t)N * HEADS;
  float* amax  = tbuf + (size_t)N * HEADS;
  float* asum  = amax + (size_t)N * HEADS;
  float* alpha = asum + (size_t)N * HEADS;

  k_gemm<<<(N + 15) / 16, 128, 0, stream>>>(x, w, hbuf, N);

  const int nh = N * HEADS;
  k_node_prep<<<(nh + 255) / 256, 256, 0, stream>>>(hbuf, att, sbuf, tbuf,
                                                    amax, asum, out, nh);
  const int eh = E * HEADS;
  k_edge_alpha<<<(eh + 255) / 256, 256, 0, stream>>>(ei, sbuf, tbuf, alpha, amax, E);
  k_edge_agg<<<(eh + 255) / 256, 256, 0, stream>>>(ei, alpha, amax, hbuf, asum, out, E);

  const int total = N * HC;
  k_finalize<<<(total + 255) / 256, 256, 0, stream>>>(out, asum, bias, total);
}